// QConv1D_72988674228631
// MI455X (gfx1250) — compile-verified
//
#include <hip/hip_runtime.h>

typedef __attribute__((ext_vector_type(2))) float v2f;
typedef __attribute__((ext_vector_type(8))) float v8f;

#define IN_CH   2
#define KW      4
#define OUT_CH  8
#define LAYERS  2
#define NQ      8
#define LIN     512
#define BATCH   16
#define LOUT    509                  // (512 - 4) + 1
#define NTOT    (BATCH * LOUT)       // 8144
#define SIMS_PER_WAVE   4
#define WAVES_PER_BLOCK 8
#define CHUNKS  ((NTOT + SIMS_PER_WAVE - 1) / SIMS_PER_WAVE)   // 2036

// RY matrix R = [[c,-s],[s,c]]; returns R[ib][jb]
__device__ __forceinline__ float ryfac(int ib, int jb, float c, float s) {
    return jb ? (ib ? c : -s) : (ib ? s : c);
}

__global__ __launch_bounds__(256) void qconv1d_wmma_kernel(
        const float* __restrict__ x, const float* __restrict__ w,
        float* __restrict__ out) {
    __shared__ float lds[WAVES_PER_BLOCK][256];

    const int lane = threadIdx.x & 31;
    const int wv   = threadIdx.x >> 5;
    const int hi   = lane >> 4;       // which half-wave (rows/K upper half)
    const int lo   = lane & 15;       // column / row index within 16
    float* sm = lds[wv];

    const int gw    = blockIdx.x * WAVES_PER_BLOCK + wv;  // global wave id
    const int o     = gw / CHUNKS;                        // output channel
    const int chunk = gw % CHUNKS;
    if (o >= OUT_CH) return;                              // wave-uniform

    // ---- Per-o weight matrices, built directly in WMMA operand layouts ----
    // pr[l][v]: Prow (16x16, row qubits 0..3) in A-layout: element (m=lo, k)
    // pc[l][v]: Pcol^T (16x16, col qubits 4..7) in B-layout: element (k, n=lo)
    // A/B 16x16x4 slice j, vgpr v: k = 4*(v>>1) + 2*hi + (v&1)
    float pr[LAYERS][8], pc[LAYERS][8];
    #pragma unroll
    for (int l = 0; l < LAYERS; ++l) {
        float cw[NQ], sw[NQ];
        #pragma unroll
        for (int q = 0; q < NQ; ++q) {
            float th = 0.5f * w[(o * NQ + q) * LAYERS + l];
            cw[q] = __cosf(th);
            sw[q] = __sinf(th);
        }
        #pragma unroll
        for (int v = 0; v < 8; ++v) {
            const int k = 4 * (v >> 1) + 2 * hi + (v & 1);
            float a = 1.f, b = 1.f;
            #pragma unroll
            for (int t = 0; t < 4; ++t) {
                // bit t of row index <-> qubit 3-t; bit t of col index <-> qubit 7-t
                a *= ryfac((lo >> t) & 1, (k >> t) & 1, cw[3 - t], sw[3 - t]);
                b *= ryfac((lo >> t) & 1, (k >> t) & 1, cw[7 - t], sw[7 - t]);
            }
            pr[l][v] = a;
            pc[l][v] = b;
        }
    }

    // Per-lane column sign helpers (c = lo)
    const int czc = __popc(lo & (lo >> 1)) & 1;   // CZ pairs inside col bits
    const int cb3 = (lo >> 3) & 1;                // b4 (col MSB) for cross pair
    const int zc  = (0x6996 >> lo) & 1;           // parity of col bits

    for (int s = 0; s < SIMS_PER_WAVE; ++s) {
        const int n = chunk * SIMS_PER_WAVE + s;
        if (n >= NTOT) break;                     // wave-uniform
        const int bx = n / LOUT;
        const int px = n % LOUT;

        // ---- angles from the conv patch; state after RYs on |0>: rank-1 ----
        float cx[NQ], sx[NQ];
        #pragma unroll
        for (int q = 0; q < NQ; ++q) {
            const int ch = q >> 2, ko = q & 3;    // q = ch*K + ko
            float th = 0.5f * x[(bx * IN_CH + ch) * LIN + px + ko];
            cx[q] = __cosf(th);
            sx[q] = __sinf(th);
        }
        float vc = 1.f;                           // v[col=lo], qubits 4..7
        #pragma unroll
        for (int t = 0; t < 4; ++t) vc *= ((lo >> t) & 1) ? sx[7 - t] : cx[7 - t];

        v8f d;                                    // state in WMMA C/D layout
        #pragma unroll
        for (int i = 0; i < 8; ++i) {
            const int r = i + 8 * hi;
            float u = 1.f;                        // u[row=r], qubits 0..3
            #pragma unroll
            for (int t = 0; t < 4; ++t) u *= ((r >> t) & 1) ? sx[3 - t] : cx[3 - t];
            d[i] = u * vc;
        }

        #pragma unroll
        for (int l = 0; l < LAYERS; ++l) {
            // ---- T = M * Pcol^T : need M in A-layout (via LDS bounce) ----
            #pragma unroll
            for (int i = 0; i < 8; ++i) sm[(i + 8 * hi) * 16 + lo] = d[i];
            float am[8];
            #pragma unroll
            for (int v = 0; v < 8; ++v) {
                const int k = 4 * (v >> 1) + 2 * hi + (v & 1);
                am[v] = sm[lo * 16 + k];          // M[m=lo][k]
            }
            v8f t = {};
            #pragma unroll
            for (int j = 0; j < 4; ++j) {
                v2f A; A.x = am[2 * j];       A.y = am[2 * j + 1];
                v2f B; B.x = pc[l][2 * j];    B.y = pc[l][2 * j + 1];
                t = __builtin_amdgcn_wmma_f32_16x16x4_f32(
                        false, A, false, B, (short)0, t, false, false);
            }
            // ---- M' = Prow * T : need T in B-layout (via LDS bounce) ----
            #pragma unroll
            for (int i = 0; i < 8; ++i) sm[(i + 8 * hi) * 16 + lo] = t[i];
            float bt[8];
            #pragma unroll
            for (int v = 0; v < 8; ++v) {
                const int k = 4 * (v >> 1) + 2 * hi + (v & 1);
                bt[v] = sm[k * 16 + lo];          // T[k][n=lo]
            }
            v8f d2 = {};
            #pragma unroll
            for (int j = 0; j < 4; ++j) {
                v2f A; A.x = pr[l][2 * j];    A.y = pr[l][2 * j + 1];
                v2f B; B.x = bt[2 * j];       B.y = bt[2 * j + 1];
                d2 = __builtin_amdgcn_wmma_f32_16x16x4_f32(
                        false, A, false, B, (short)0, d2, false, false);
            }
            // ---- CZ ladder sign, elementwise in D-layout ----
            #pragma unroll
            for (int i = 0; i < 8; ++i) {
                const int r = i + 8 * hi;
                const int sgn = (__popc(r & (r >> 1)) + czc + ((r & 1) & cb3)) & 1;
                d[i] = sgn ? -d2[i] : d2[i];
            }
        }

        // ---- <Z...Z> = sum parity(s) * amp^2 ----
        float acc = 0.f;
        #pragma unroll
        for (int i = 0; i < 8; ++i) {
            const int r = i + 8 * hi;
            const int sgn = ((0x6996 >> r) & 1) ^ zc;
            const float v2 = d[i] * d[i];
            acc += sgn ? -v2 : v2;
        }
        #pragma unroll
        for (int off = 16; off >= 1; off >>= 1)
            acc += __shfl_xor(acc, off, 32);

        if (lane == 0) {
            // sim index i = o*NTOT + n ; reference reshapes (O*N,) -> (B,L,O)
            const int i  = o * NTOT + n;
            const int bb = i / (LOUT * OUT_CH);
            const int rm = i % (LOUT * OUT_CH);
            const int pp = rm / OUT_CH;
            const int oo = rm % OUT_CH;
            out[(bb * OUT_CH + oo) * LOUT + pp] = acc;
        }
    }
}

extern "C" void kernel_launch(void* const* d_in, const int* in_sizes, int n_in,
                              void* d_out, int out_size, void* d_ws, size_t ws_size,
                              hipStream_t stream) {
    const float* x = (const float*)d_in[0];   // (16, 2, 512) f32
    const float* w = (const float*)d_in[1];   // (8, 8, 2) f32
    float* out = (float*)d_out;               // (16, 8, 509) f32

    const int total_waves = OUT_CH * CHUNKS;                       // 16288
    const int blocks = (total_waves + WAVES_PER_BLOCK - 1) / WAVES_PER_BLOCK;
    qconv1d_wmma_kernel<<<dim3(blocks), dim3(256), 0, stream>>>(x, w, out);
}